// TritonRMSNormW8A8Impl_78872779424012
// MI455X (gfx1250) — compile-verified
//
#include <hip/hip_runtime.h>
#include <hip/hip_fp16.h>
#include <cstdint>
#include <cstddef>

// Fused residual-add + RMSNorm + dynamic per-token int8 quant for MI455X.
//
// Memory-bound kernel: 7 B/element of compulsory HBM traffic (~940 MB total)
// => ~40us floor at 23.3 TB/s. Design: single pass over x/res (h kept in
// registers across both reductions), B128 non-temporal vector accesses for the
// streaming tensors, and the CDNA5 async global->LDS path (ASYNCcnt) for the
// L2-resident weight vector.
//
// Output layout in d_out (byte-packed, reference return order):
//   [0,                T*H)            : q            int8
//   [T*H,              T*H + 4T)       : scale        float32
//   [T*H + 4T,         T*H + 4T + 2TH) : residual_out float16

typedef uint32_t u32x4 __attribute__((ext_vector_type(4)));

#define EPS_RMS 1e-6f
#define QMAX_F  127.0f

static __device__ __forceinline__ float2 h2u_to_f2(uint32_t u) {
    __half2 h;
    __builtin_memcpy(&h, &u, sizeof(h));
    return __half22float2(h);
}

static __device__ __forceinline__ uint32_t f2_to_h2u(float a, float b) {
    __half2 h = __floats2half2_rn(a, b);
    uint32_t u;
    __builtin_memcpy(&u, &h, sizeof(u));
    return u;
}

// One 256-thread block (8 wave32) per token row; 16 elements per thread.
// Requires H == 4096.
__global__ __launch_bounds__(256)
void rmsnorm_w8a8_fused(const __half* __restrict__ x,
                        const __half* __restrict__ residual,
                        const float*  __restrict__ weight,
                        int8_t*       __restrict__ q_out,
                        float*        __restrict__ scale_out,
                        __half*       __restrict__ res_out,
                        int H)
{
    const int row  = blockIdx.x;
    const int tid  = threadIdx.x;
    const int lane = tid & 31;   // wave32
    const int wave = tid >> 5;   // 8 waves per block

    __shared__ float w_lds[4096];   // 16 KB weight tile
    __shared__ float red_ss[8];
    __shared__ float red_mx[8];

    // --- Phase 0: async-copy this thread's own 64B weight slice global->LDS.
    // CDNA5 GLOBAL_LOAD_ASYNC_TO_LDS_B128 (GVS mode: saddr base + vgpr offset),
    // tracked by ASYNCcnt. Each thread later reads back only the bytes it
    // issued itself, so a per-wave s_wait_asynccnt is sufficient (no barrier).
    {
        const uint32_t lds_base = (uint32_t)(uintptr_t)(&w_lds[0]); // low 32b = LDS offset
        const uint32_t lds_off  = lds_base + (uint32_t)tid * 64u;
        const uint32_t g_off    = (uint32_t)tid * 64u;
#pragma unroll
        for (int i = 0; i < 4; ++i) {
            uint32_t l = lds_off + 16u * (uint32_t)i;
            uint32_t g = g_off   + 16u * (uint32_t)i;
            asm volatile("global_load_async_to_lds_b128 %0, %1, %2"
                         :
                         : "v"(l), "v"(g), "s"(weight)
                         : "memory");
        }
    }

    // --- Phase 1: stream x/res (non-temporal B128), h = x + res in fp32 regs,
    // write residual_out immediately.
    const size_t base = (size_t)row * (size_t)H;
    const u32x4* xv = reinterpret_cast<const u32x4*>(x        + base) + (size_t)tid * 2;
    const u32x4* rv = reinterpret_cast<const u32x4*>(residual + base) + (size_t)tid * 2;

    u32x4 xa = __builtin_nontemporal_load(xv);
    u32x4 xb = __builtin_nontemporal_load(xv + 1);
    u32x4 ra = __builtin_nontemporal_load(rv);
    u32x4 rb = __builtin_nontemporal_load(rv + 1);

    float h[16];
    u32x4 oa, ob;
#pragma unroll
    for (int k = 0; k < 4; ++k) {
        float2 fx = h2u_to_f2(xa[k]);
        float2 fr = h2u_to_f2(ra[k]);
        h[2 * k]     = fx.x + fr.x;
        h[2 * k + 1] = fx.y + fr.y;
        oa[k] = f2_to_h2u(h[2 * k], h[2 * k + 1]);
    }
#pragma unroll
    for (int k = 0; k < 4; ++k) {
        float2 fx = h2u_to_f2(xb[k]);
        float2 fr = h2u_to_f2(rb[k]);
        h[8 + 2 * k]     = fx.x + fr.x;
        h[8 + 2 * k + 1] = fx.y + fr.y;
        ob[k] = f2_to_h2u(h[8 + 2 * k], h[8 + 2 * k + 1]);
    }

    u32x4* ro = reinterpret_cast<u32x4*>(res_out + base) + (size_t)tid * 2;
    __builtin_nontemporal_store(oa, ro);
    __builtin_nontemporal_store(ob, ro + 1);

    // --- Phase 2: sum-of-squares reduction (wave32 shfl_xor + LDS cross-wave).
    float ss = 0.f;
#pragma unroll
    for (int k = 0; k < 16; ++k) ss = fmaf(h[k], h[k], ss);
#pragma unroll
    for (int m = 16; m >= 1; m >>= 1)
        ss += __shfl_xor(ss, m, 32);
    if (lane == 0) red_ss[wave] = ss;
    __syncthreads();
    float tot = 0.f;
#pragma unroll
    for (int i = 0; i < 8; ++i) tot += red_ss[i];   // broadcast reads, no conflicts

    const float rstd = rsqrtf(tot * (1.0f / (float)H) + EPS_RMS);

    // --- Phase 3: weight from LDS (async copy must have landed), y = h*rstd*w.
    asm volatile("s_wait_asynccnt 0" ::: "memory");

    float y[16];
    float amax = 0.f;
    {
        const float4* wl = reinterpret_cast<const float4*>(&w_lds[tid * 16]);
#pragma unroll
        for (int k = 0; k < 4; ++k) {
            float4 w4 = wl[k];            // ds_load_b128
            y[4 * k]     = h[4 * k]     * rstd * w4.x;
            y[4 * k + 1] = h[4 * k + 1] * rstd * w4.y;
            y[4 * k + 2] = h[4 * k + 2] * rstd * w4.z;
            y[4 * k + 3] = h[4 * k + 3] * rstd * w4.w;
        }
    }
#pragma unroll
    for (int k = 0; k < 16; ++k) amax = fmaxf(amax, fabsf(y[k]));

    // --- Phase 4: max-abs reduction for dynamic per-token scale.
#pragma unroll
    for (int m = 16; m >= 1; m >>= 1)
        amax = fmaxf(amax, __shfl_xor(amax, m, 32));
    if (lane == 0) red_mx[wave] = amax;
    __syncthreads();
    float gmax = red_mx[0];
#pragma unroll
    for (int i = 1; i < 8; ++i) gmax = fmaxf(gmax, red_mx[i]);

    if (tid == 0) scale_out[row] = gmax * (1.0f / QMAX_F);

    // --- Phase 5: quantize (RNE like jnp.round), pack 16 int8, one B128 store.
    const float inv = QMAX_F / gmax;
    u32x4 qp;
#pragma unroll
    for (int k = 0; k < 4; ++k) {
        uint32_t b = 0;
#pragma unroll
        for (int j = 0; j < 4; ++j) {
            float v = rintf(y[4 * k + j] * inv);
            v = fminf(fmaxf(v, -128.f), 127.f);
            int iv = (int)v;
            b |= ((uint32_t)iv & 0xffu) << (8 * j);
        }
        qp[k] = b;
    }
    u32x4* qo = reinterpret_cast<u32x4*>(q_out + base) + tid;
    __builtin_nontemporal_store(qp, qo);
}

extern "C" void kernel_launch(void* const* d_in, const int* in_sizes, int n_in,
                              void* d_out, int out_size, void* d_ws, size_t ws_size,
                              hipStream_t stream) {
    (void)n_in; (void)out_size; (void)d_ws; (void)ws_size;

    const __half* x        = (const __half*)d_in[0];
    const __half* residual = (const __half*)d_in[1];
    const float*  weight   = (const float*) d_in[2];

    const int H = in_sizes[2];              // 4096
    const int T = in_sizes[0] / H;          // 32768

    int8_t* q_out     = (int8_t*)d_out;
    float*  scale_out = (float*)((char*)d_out + (size_t)T * (size_t)H);
    __half* res_out   = (__half*)((char*)d_out + (size_t)T * (size_t)H
                                               + (size_t)T * sizeof(float));

    // Kernel assumes H == 256 threads * 16 elements.
    if (H != 4096) return;

    rmsnorm_w8a8_fused<<<T, 256, 0, stream>>>(x, residual, weight,
                                              q_out, scale_out, res_out, H);
}